// SAGENet_14362370638305
// MI455X (gfx1250) — compile-verified
//
#include <hip/hip_runtime.h>
#include <hip/hip_bf16.h>

typedef __attribute__((ext_vector_type(2))) float v2f;
typedef __attribute__((ext_vector_type(8))) float v8f;

#define D 128
#define NEG_SLOPE 0.01f

// ---------------------------------------------------------------- utilities
__global__ void zero_kernel(float* __restrict__ p, int n) {
    int i = blockIdx.x * blockDim.x + threadIdx.x;
    if (i < n) p[i] = 0.0f;
}

// deg[dst[e]] += 1  (degrees are identical for every layer -> computed once)
__global__ void degree_kernel(const int* __restrict__ dst, float* __restrict__ deg, int nE) {
    int i = blockIdx.x * blockDim.x + threadIdx.x;
    if (i < nE) atomicAdd(&deg[dst[i]], 1.0f);
}

__global__ void invdeg_kernel(float* __restrict__ deg, int n) {
    int i = blockIdx.x * blockDim.x + threadIdx.x;
    if (i < n) deg[i] = 1.0f / fmaxf(deg[i], 1.0f);
}

// ------------------------------------------------- scatter-add of neighbor rows
// one wave per edge; lane L moves floats [4L..4L+3] of the 128-wide row
__global__ __launch_bounds__(256) void scatter_kernel(
    const float* __restrict__ x, const int* __restrict__ src,
    const int* __restrict__ dst, float* __restrict__ agg, int nE)
{
    int wid  = (blockIdx.x * blockDim.x + threadIdx.x) >> 5;
    int lane = threadIdx.x & 31;
    if (wid >= nE) return;
    int s = src[wid];
    int d = dst[wid];
    float4 v = ((const float4*)(x + (size_t)s * D))[lane];
    float* o = agg + (size_t)d * D + lane * 4;
    atomicAdd(o + 0, v.x);
    atomicAdd(o + 1, v.y);
    atomicAdd(o + 2, v.z);
    atomicAdd(o + 3, v.w);
}

// ------------------------------------------------- fused SAGEConv layer GEMM
// xout = act( (agg*invdeg) @ Wl + xin @ Wr + b )
// block = 256 threads = 8 waves; wave owns 16 rows x all 128 cols.
// Both weight matrices staged once in 128KB LDS, pair-contiguous layout
// [p][k/2][n]{k,k+1} so every B fragment is a single aligned ds_load_b64.
__global__ __launch_bounds__(256) void sage_gemm_kernel(
    const float* __restrict__ xin, const float* __restrict__ agg,
    const float* __restrict__ invdeg,
    const float* __restrict__ Wl, const float* __restrict__ Wr,
    const float* __restrict__ bias,
    float* __restrict__ xout, int M, int act)
{
    __shared__ v2f wlds[2 * 64 * 128];            // 128 KB: [p][kpair][n] -> {W[2kp][n], W[2kp+1][n]}

    const int lane    = threadIdx.x & 31;
    const int wave    = threadIdx.x >> 5;
    const int l15     = lane & 15;
    const int halfsel = lane >> 4;                // 0: K {k,k+1}; 1: K {k+2,k+3}
    const int m0      = (blockIdx.x * 8 + wave) * 16;
    const bool active = (m0 < M);                 // wave-uniform -> EXEC all-1s inside
    const int arow    = active ? (m0 + l15) : 0;
    const float idg   = invdeg[arow];

    v8f acc[8] = {};

    // ---- stage Wl and Wr (pair-contiguous), fully coalesced -------------
    // pair index idx = p*8192 + kp*128 + n
    #pragma unroll 4
    for (int it = 0; it < 64; ++it) {
        const int idx = threadIdx.x + it * 256;
        const int n   = idx & 127;
        const int kp  = (idx >> 7) & 63;
        const int p   = idx >> 13;
        const float* W = p ? Wr : Wl;
        v2f v;
        v.x = W[(2 * kp)     * D + n];
        v.y = W[(2 * kp + 1) * D + n];
        wlds[idx] = v;
    }
    __syncthreads();

    if (active) {
        const v2f* aRow = (const v2f*)(agg + (size_t)arow * D);
        const v2f* xRow = (const v2f*)(xin + (size_t)arow * D);

        // ---- phase 1: (agg * invdeg) @ Wl -------------------------------
        for (int k = 0; k < D; k += 4) {
            const int kp = (k >> 1) + halfsel;    // even-K pair index
            v2f a = aRow[kp];
            a.x *= idg;
            a.y *= idg;
            const v2f* wp = wlds + kp * 128 + l15;
            #pragma unroll
            for (int t = 0; t < 8; ++t) {
                acc[t] = __builtin_amdgcn_wmma_f32_16x16x4_f32(
                    false, a, false, wp[t * 16], (short)0, acc[t], false, false);
            }
        }
        // ---- phase 2: xin @ Wr ------------------------------------------
        for (int k = 0; k < D; k += 4) {
            const int kp = (k >> 1) + halfsel;
            const v2f a = xRow[kp];
            const v2f* wp = wlds + 8192 + kp * 128 + l15;
            #pragma unroll
            for (int t = 0; t < 8; ++t) {
                acc[t] = __builtin_amdgcn_wmma_f32_16x16x4_f32(
                    false, a, false, wp[t * 16], (short)0, acc[t], false, false);
            }
        }

        // ---- epilogue: bias + leaky_relu, C/D layout per ISA ------------
        const int r0 = m0 + halfsel * 8;          // VGPR r holds rows r / r+8
        #pragma unroll
        for (int t = 0; t < 8; ++t) {
            const int n  = t * 16 + l15;
            const float bn = bias[n];
            #pragma unroll
            for (int r = 0; r < 8; ++r) {
                float v = acc[t][r] + bn;
                if (act) v = (v > 0.0f) ? v : v * NEG_SLOPE;
                xout[(size_t)(r0 + r) * D + n] = v;
            }
        }
    }
}

// ------------------------------------------------- edge scoring: dot products
// one wave per label edge: lane L handles float4 chunk L, then wave32 reduce
__global__ __launch_bounds__(256) void edge_dot_kernel(
    const float* __restrict__ x, const int* __restrict__ ia,
    const int* __restrict__ ib, float* __restrict__ out, int n)
{
    int wid  = (blockIdx.x * blockDim.x + threadIdx.x) >> 5;
    int lane = threadIdx.x & 31;
    if (wid >= n) return;
    int a = ia[wid], b = ib[wid];
    float4 va = ((const float4*)(x + (size_t)a * D))[lane];
    float4 vb = ((const float4*)(x + (size_t)b * D))[lane];
    float s = va.x * vb.x + va.y * vb.y + va.z * vb.z + va.w * vb.w;
    #pragma unroll
    for (int off = 16; off > 0; off >>= 1)
        s += __shfl_xor(s, off, 32);
    if (lane == 0) out[wid] = s;
}

// ---------------------------------------------------------------- launcher
extern "C" void kernel_launch(void* const* d_in, const int* in_sizes, int n_in,
                              void* d_out, int out_size, void* d_ws, size_t ws_size,
                              hipStream_t stream) {
    const float* x0   = (const float*)d_in[0];   // [N,128]
    const int*   ei   = (const int*)  d_in[1];   // [2,nE]
    const int*   eli  = (const int*)  d_in[2];   // [2,nL]
    const float* Wl   = (const float*)d_in[3];   // [8,128,128]
    const float* Wr   = (const float*)d_in[4];   // [8,128,128]
    const float* bias = (const float*)d_in[5];   // [8,128]
    float*       out  = (float*)d_out;

    const int nNodes = in_sizes[0] / D;
    const int nEdges = in_sizes[1] / 2;
    const int nLabel = in_sizes[2] / 2;

    const int* src = ei;
    const int* dst = ei + nEdges;

    // workspace: xbuf0 | xbuf1 | agg | invdeg  (~77 MB)
    float* xbuf0  = (float*)d_ws;
    float* xbuf1  = xbuf0 + (size_t)nNodes * D;
    float* agg    = xbuf1 + (size_t)nNodes * D;
    float* invdeg = agg   + (size_t)nNodes * D;

    // degrees (layer-invariant): deg -> 1/max(deg,1)
    zero_kernel   <<<(nNodes + 255) / 256, 256, 0, stream>>>(invdeg, nNodes);
    degree_kernel <<<(nEdges + 255) / 256, 256, 0, stream>>>(dst, invdeg, nEdges);
    invdeg_kernel <<<(nNodes + 255) / 256, 256, 0, stream>>>(invdeg, nNodes);

    const int aggN       = nNodes * D;
    const int gemmBlocks = (nNodes + 127) / 128;   // 8 waves x 16 rows per block

    const float* xin  = x0;
    float*       xout = xbuf0;
    for (int layer = 0; layer < 8; ++layer) {
        zero_kernel   <<<(aggN + 255) / 256, 256, 0, stream>>>(agg, aggN);
        scatter_kernel<<<(nEdges + 7) / 8,   256, 0, stream>>>(xin, src, dst, agg, nEdges);
        sage_gemm_kernel<<<gemmBlocks, 256, 0, stream>>>(
            xin, agg, invdeg,
            Wl + (size_t)layer * D * D,
            Wr + (size_t)layer * D * D,
            bias + (size_t)layer * D,
            xout, nNodes, (layer < 6) ? 1 : 0);
        xin  = xout;
        xout = (xout == xbuf0) ? xbuf1 : xbuf0;
    }

    edge_dot_kernel<<<(nLabel + 7) / 8, 256, 0, stream>>>(xin, eli, eli + nLabel, out, nLabel);
}